// HungarianMatcher_80496277061948
// MI455X (gfx1250) — compile-verified
//
#include <hip/hip_runtime.h>
#include <hip/hip_bf16.h>

// DETR HungarianMatcher cost matrix on gfx1250 (MI455X, wave32).
//
//   C[q,t] = L1(pred_box[q], tgt_box[t]) - softmax(logits)[q, label[t]] - GIoU(q,t)
//
// Shapes: bs=32, nq=300, nc=92, nt=50 -> Q=9600, T=1600, out = [9600,1600] f32.
// Class cost is computed as a one-hot GEMM P[16,96] @ O[96,16] per tile using
// V_WMMA_F32_16X16X4_F32 (matrix pipe, co-executes with the VALU L1/GIoU
// epilogue). All target data is staged/preconverted in LDS so the inner loop
// has no global-memory waits.

typedef __attribute__((ext_vector_type(2))) float v2f;
typedef __attribute__((ext_vector_type(8))) float v8f;

#define NC        92
#define NCP       96            // padded class count (one-hot rows 92..95 always 0)
#define KSTEPS    (NCP / 4)     // 24 WMMA k-steps
#define QTOT      9600
#define TTOT      1600
#define TTILES    (TTOT / 16)   // 100
#define TT_PER_BY (TTILES / 4)  // 25 t-tiles per blockIdx.y slice
#define T_PER_BY  (TT_PER_BY * 16)  // 400 targets staged per block

__global__ __launch_bounds__(256)
void hungarian_cost_kernel(const float* __restrict__ logits,      // [9600,92]
                           const float* __restrict__ pred_boxes,  // [9600,4] cxcywh
                           const int*   __restrict__ tgt_labels,  // [1600]
                           const float* __restrict__ tgt_boxes,   // [1600,4] cxcywh
                           float*       __restrict__ out)         // [9600,1600]
{
    __shared__ __align__(16) float Plds[16][NCP];      // softmax probs, 16 q-rows
    __shared__ __align__(16) float Blds[16][12];       // per-q: cxcywh, xyxy, area
    __shared__ __align__(16) float Tlds[T_PER_BY][12]; // per-t: cxcywh, xyxy, area, cls

    const int tid   = threadIdx.x;
    const int lane  = tid & 31;
    const int wv    = tid >> 5;            // wave 0..7
    const int qbase = blockIdx.x * 16;
    const int ttlo  = blockIdx.y * TT_PER_BY;

    // ---------- Phase 0a: softmax of 16 logit rows (one wave per 2 rows) ----------
    #pragma unroll
    for (int rsub = 0; rsub < 2; ++rsub) {
        const int rr = wv * 2 + rsub;
        const float* row = logits + (size_t)(qbase + rr) * NC;
        float v0 = row[lane];
        float v1 = row[lane + 32];
        const bool has2 = lane < (NC - 64);            // lanes 0..27 cover c=64..91
        float v2 = has2 ? row[lane + 64] : -__builtin_inff();
        float m = fmaxf(fmaxf(v0, v1), v2);
        #pragma unroll
        for (int off = 16; off >= 1; off >>= 1)
            m = fmaxf(m, __shfl_xor(m, off, 32));
        float e0 = __expf(v0 - m);
        float e1 = __expf(v1 - m);
        float e2 = has2 ? __expf(v2 - m) : 0.0f;
        float s = e0 + e1 + e2;
        #pragma unroll
        for (int off = 16; off >= 1; off >>= 1)
            s += __shfl_xor(s, off, 32);
        const float inv = __builtin_amdgcn_rcpf(s);
        Plds[rr][lane]      = e0 * inv;
        Plds[rr][lane + 32] = e1 * inv;
        if (has2) Plds[rr][lane + 64] = e2 * inv;
        if (lane < NCP - NC) Plds[rr][NC + lane] = 0.0f;   // zero class padding
    }

    // ---------- Phase 0b: pred box precompute (cxcywh -> xyxy, area) ----------
    if (tid < 16) {
        const float* pb = pred_boxes + (size_t)(qbase + tid) * 4;
        const float cx = pb[0], cy = pb[1], w = pb[2], h = pb[3];
        const float x0 = cx - 0.5f * w, y0 = cy - 0.5f * h;
        const float x1 = cx + 0.5f * w, y1 = cy + 0.5f * h;
        Blds[tid][0] = cx; Blds[tid][1] = cy; Blds[tid][2] = w;  Blds[tid][3] = h;
        Blds[tid][4] = x0; Blds[tid][5] = y0; Blds[tid][6] = x1; Blds[tid][7] = y1;
        Blds[tid][8] = (x1 - x0) * (y1 - y0);
    }

    // ---------- Phase 0c: stage + preconvert this slice's 400 targets ----------
    for (int j = tid; j < T_PER_BY; j += 256) {
        const int t = ttlo * 16 + j;
        const float4 tb = *(const float4*)(tgt_boxes + (size_t)t * 4);
        const int cls = tgt_labels[t];
        const float tx0 = tb.x - 0.5f * tb.z, ty0 = tb.y - 0.5f * tb.w;
        const float tx1 = tb.x + 0.5f * tb.z, ty1 = tb.y + 0.5f * tb.w;
        Tlds[j][0] = tb.x; Tlds[j][1] = tb.y; Tlds[j][2] = tb.z; Tlds[j][3] = tb.w;
        Tlds[j][4] = tx0;  Tlds[j][5] = ty0;  Tlds[j][6] = tx1;  Tlds[j][7] = ty1;
        Tlds[j][8] = (tx1 - tx0) * (ty1 - ty0);
        *(int*)&Tlds[j][9] = cls;
    }
    __syncthreads();

    // ---------- A fragments: 16x96 prob tile, ISA 16x4 f32 A layout ----------
    // lanes 0-15: M=lane, VGPR0/1 = K+0/K+1 ; lanes 16-31: M=lane-16, K+2/K+3
    const int n  = lane & 15;    // A-row M / B-col N / target column within tile
    const int hi = lane >> 4;
    v2f afrag[KSTEPS];
    #pragma unroll
    for (int k = 0; k < KSTEPS; ++k) {
        const int c0 = 4 * k + 2 * hi;
        afrag[k].x = Plds[n][c0];
        afrag[k].y = Plds[n][c0 + 1];
    }

    // ---------- Phase 1: loop over 16-wide t-tiles (LDS-only inner loop) ----------
    for (int tt = wv; tt < TT_PER_BY; tt += 8) {
        const int j = tt * 16 + n;                 // index into Tlds
        const float* tp = &Tlds[j][0];
        const float tcx = tp[0], tcy = tp[1], tw = tp[2], th = tp[3];
        const float tx0 = tp[4], ty0 = tp[5], tx1 = tp[6], ty1 = tp[7];
        const float tarea = tp[8];
        const int   cls  = *(const int*)&tp[9];

        // one-hot helper: lane's label hits exactly one k-step / slot pair
        const int   khot = cls >> 2;
        const int   slot = cls & 3;
        const float e0 = (slot == 2 * hi)     ? 1.0f : 0.0f;
        const float e1 = (slot == 2 * hi + 1) ? 1.0f : 0.0f;

        // ---- class cost via one-hot GEMM on the matrix pipe ----
        v8f acc = {};
        #pragma unroll
        for (int k = 0; k < KSTEPS; ++k) {
            const bool hit = (k == khot);          // 1 v_cmp per step
            v2f b;
            b.x = hit ? e0 : 0.0f;                 // 2 v_cndmask per step
            b.y = hit ? e1 : 0.0f;
            acc = __builtin_amdgcn_wmma_f32_16x16x4_f32(
                false, afrag[k], false, b, (short)0, acc, false, false);
        }
        // acc[r] = softmax(logits)[qbase + r + 8*hi, cls]

        // ---- L1 + GIoU epilogue on the vector pipe (co-executes with WMMA) ----
        #pragma unroll
        for (int r = 0; r < 8; ++r) {
            const int ql = r + 8 * hi;
            const float* bx = &Blds[ql][0];
            const float cx = bx[0], cy = bx[1], w = bx[2], h = bx[3];
            const float x0 = bx[4], y0 = bx[5], x1 = bx[6], y1 = bx[7];
            const float area1 = bx[8];

            const float l1 = fabsf(cx - tcx) + fabsf(cy - tcy)
                           + fabsf(w  - tw)  + fabsf(h  - th);

            const float ix0 = fmaxf(x0, tx0), iy0 = fmaxf(y0, ty0);
            const float ix1 = fminf(x1, tx1), iy1 = fminf(y1, ty1);
            const float iw = fmaxf(ix1 - ix0, 0.0f), ih = fmaxf(iy1 - iy0, 0.0f);
            const float inter = iw * ih;
            const float uni = area1 + tarea - inter;

            const float ex0 = fminf(x0, tx0), ey0 = fminf(y0, ty0);
            const float ex1 = fmaxf(x1, tx1), ey1 = fmaxf(y1, ty1);
            const float ew = fmaxf(ex1 - ex0, 0.0f), eh = fmaxf(ey1 - ey0, 0.0f);
            const float areae = ew * eh;

            const float iou  = inter * __builtin_amdgcn_rcpf(uni);
            const float giou = iou - (areae - uni) * __builtin_amdgcn_rcpf(areae);

            out[(size_t)(qbase + ql) * TTOT + (ttlo + tt) * 16 + n] = l1 - acc[r] - giou;
        }
    }
}

extern "C" void kernel_launch(void* const* d_in, const int* in_sizes, int n_in,
                              void* d_out, int out_size, void* d_ws, size_t ws_size,
                              hipStream_t stream) {
    const float* logits  = (const float*)d_in[0];  // pred_logits [32,300,92]
    const float* pboxes  = (const float*)d_in[1];  // pred_boxes  [32,300,4]
    const int*   tlabels = (const int*)  d_in[2];  // tgt_labels  [32,50]
    const float* tboxes  = (const float*)d_in[3];  // tgt_boxes   [32,50,4]
    float* out = (float*)d_out;                    // [32,300,1600] f32

    dim3 grid(QTOT / 16, 4);   // 600 q-tiles x 4 t-slices = 2400 blocks, 8 waves each
    hungarian_cost_kernel<<<grid, 256, 0, stream>>>(logits, pboxes, tlabels, tboxes, out);
}